// SparseLocalSelfAttention_25821343384101
// MI455X (gfx1250) — compile-verified
//
#include <hip/hip_runtime.h>
#include <math.h>

#define NPTS 20000
#define CCH  256
#define HH   2
#define EE   300000
#define QKVLD 768

typedef float v2f __attribute__((ext_vector_type(2)));
typedef float v8f __attribute__((ext_vector_type(8)));

// Guaranteed hardware fp32 atomic add (L2-side RMW, device scope, no return).
__device__ __forceinline__ void atomicAddF32(float* p, float v) {
    asm volatile("global_atomic_add_f32 %0, %1, off scope:SCOPE_DEV"
                 :: "v"(p), "v"(v) : "memory");
}

__device__ __forceinline__ void atomicMaxF(int* addr, float v) {
    if (v >= 0.0f) atomicMax(addr, __float_as_int(v));
    else           atomicMin((unsigned int*)addr, (unsigned int)__float_as_int(v));
}

// ---------------------------------------------------------------- init scratch
__global__ void init_ws(float* __restrict__ out_raw, float* __restrict__ denom,
                        int* __restrict__ segmax) {
    int i = blockIdx.x * blockDim.x + threadIdx.x;
    if (i < NPTS * CCH) out_raw[i] = 0.0f;
    if (i < NPTS * HH) { denom[i] = 0.0f; segmax[i] = (int)0xFF800000; } // -inf
}

// ------------------------------------------------- fp32 WMMA GEMM: C = A @ W^T
// A: [M x 256] row-major, W: [Nout x 256] row-major (B = W^T, K=256).
// One wave computes a 16x64 strip: 4 accumulators share each A fragment
// (V_WMMA_F32_16X16X4_F32, 64 k-steps). Manual depth-1 software pipeline;
// __launch_bounds__(256,1) relaxes the VGPR budget so the scheduler keeps the
// prefetched fragments in distinct registers (deep loads in flight) instead of
// collapsing them into one buffer with s_wait_loadcnt 0.
// A frag (16x4): lane l, reg r -> A[M = l&15][K = 2*(l>>4)+r]   (ISA 7.12.2)
// B frag (4x16): lane l, reg r -> B[K = 2*(l>>4)+r][N = l&15] = W[l&15][K]
// C/D (16x16):   reg g, lane l -> C[M = 8*(l>>4)+g][N = l&15]
__global__ void __launch_bounds__(256, 1)
gemm_xt_wmma(const float* __restrict__ A, const float* __restrict__ W,
             float* __restrict__ Cmat, const float* __restrict__ residual,
             int mtiles, int ngroups, int ldc, int ccol0) {
    int wid  = (blockIdx.x * blockDim.x + threadIdx.x) >> 5;
    int lane = threadIdx.x & 31;
    if (wid >= mtiles * ngroups) return;
    int tm = wid / ngroups, tg = wid % ngroups;   // strip = 4 n-tiles
    int half = lane >> 4, lr = lane & 15;

    // base pointers shifted so v2f index 2*k picks floats [4k+2*half, 4k+2*half+1]
    const v2f* arow  = (const v2f*)(A + (size_t)(tm * 16 + lr) * CCH + 2 * half);
    const v2f* brow0 = (const v2f*)(W + (size_t)((tg * 4 + 0) * 16 + lr) * CCH + 2 * half);
    const v2f* brow1 = (const v2f*)(W + (size_t)((tg * 4 + 1) * 16 + lr) * CCH + 2 * half);
    const v2f* brow2 = (const v2f*)(W + (size_t)((tg * 4 + 2) * 16 + lr) * CCH + 2 * half);
    const v2f* brow3 = (const v2f*)(W + (size_t)((tg * 4 + 3) * 16 + lr) * CCH + 2 * half);

    v8f acc0 = {}, acc1 = {}, acc2 = {}, acc3 = {};

    v2f a  = arow[0];
    v2f b0 = brow0[0], b1 = brow1[0], b2 = brow2[0], b3 = brow3[0];

#pragma unroll 9
    for (int k = 0; k < CCH / 4 - 1; ++k) {
        // prefetch next k-step into fresh registers (issues before the WMMAs)
        v2f an  = arow[2 * (k + 1)];
        v2f b0n = brow0[2 * (k + 1)];
        v2f b1n = brow1[2 * (k + 1)];
        v2f b2n = brow2[2 * (k + 1)];
        v2f b3n = brow3[2 * (k + 1)];

        acc0 = __builtin_amdgcn_wmma_f32_16x16x4_f32(false, a, false, b0, (short)0, acc0, false, false);
        acc1 = __builtin_amdgcn_wmma_f32_16x16x4_f32(false, a, false, b1, (short)0, acc1, false, false);
        acc2 = __builtin_amdgcn_wmma_f32_16x16x4_f32(false, a, false, b2, (short)0, acc2, false, false);
        acc3 = __builtin_amdgcn_wmma_f32_16x16x4_f32(false, a, false, b3, (short)0, acc3, false, false);

        a = an; b0 = b0n; b1 = b1n; b2 = b2n; b3 = b3n;
    }
    acc0 = __builtin_amdgcn_wmma_f32_16x16x4_f32(false, a, false, b0, (short)0, acc0, false, false);
    acc1 = __builtin_amdgcn_wmma_f32_16x16x4_f32(false, a, false, b1, (short)0, acc1, false, false);
    acc2 = __builtin_amdgcn_wmma_f32_16x16x4_f32(false, a, false, b2, (short)0, acc2, false, false);
    acc3 = __builtin_amdgcn_wmma_f32_16x16x4_f32(false, a, false, b3, (short)0, acc3, false, false);

    v8f accs[4] = {acc0, acc1, acc2, acc3};
#pragma unroll
    for (int t = 0; t < 4; ++t) {
        int col = ccol0 + (tg * 4 + t) * 16 + lr;
#pragma unroll
        for (int g = 0; g < 8; ++g) {
            int row = tm * 16 + 8 * half + g;
            size_t oidx = (size_t)row * ldc + col;
            float v = accs[t][g];
            if (residual) v += residual[oidx];
            Cmat[oidx] = v;
        }
    }
}

// ------------------------------------------------------- per-edge attn scores
// one wave per edge; lanes 0-15: head0 dims 0-127, lanes 16-31: head1 dims 128-255
__global__ void __launch_bounds__(256)
edge_scores(const float* __restrict__ QKV, const float* __restrict__ coords,
            const float* __restrict__ Wpos, const float* __restrict__ bpos,
            const int* __restrict__ qidx, const int* __restrict__ nidx,
            float* __restrict__ scores, int* __restrict__ segmax) {
    int e    = (blockIdx.x * blockDim.x + threadIdx.x) >> 5;
    int lane = threadIdx.x & 31;
    if (e >= EE) return;
    int qi = qidx[e], ni = nidx[e];

    const float4* qp = (const float4*)(QKV + (size_t)qi * QKVLD + lane * 8);
    const float4* kp = (const float4*)(QKV + (size_t)ni * QKVLD + 256 + lane * 8);
    float4 q0 = qp[0], q1 = qp[1], k0 = kp[0], k1 = kp[1];
    float s = q0.x * k0.x + q0.y * k0.y + q0.z * k0.z + q0.w * k0.w
            + q1.x * k1.x + q1.y * k1.y + q1.z * k1.z + q1.w * k1.w;
    s += __shfl_xor(s, 1);
    s += __shfl_xor(s, 2);
    s += __shfl_xor(s, 4);
    s += __shfl_xor(s, 8);

    if ((lane & 15) == 0) {
        int h = lane >> 4;
        float rx = coords[qi * 3 + 0] - coords[ni * 3 + 0];
        float ry = coords[qi * 3 + 1] - coords[ni * 3 + 1];
        float rz = coords[qi * 3 + 2] - coords[ni * 3 + 2];
        float pos = rx * Wpos[h * 3 + 0] + ry * Wpos[h * 3 + 1]
                  + rz * Wpos[h * 3 + 2] + bpos[h];
        float sc = s * 0.08838834764831845f + pos;   // D^-0.5, D=128
        scores[e * 2 + h] = sc;
        atomicMaxF(&segmax[qi * 2 + h], sc);
    }
}

// --------------------------------------------- per-edge exp + scatter of ex*v
__global__ void __launch_bounds__(256)
edge_accum(const float* __restrict__ QKV, const float* __restrict__ scores,
           const int* __restrict__ segmax, const int* __restrict__ qidx,
           const int* __restrict__ nidx, float* __restrict__ denom,
           float* __restrict__ out_raw) {
    int e    = (blockIdx.x * blockDim.x + threadIdx.x) >> 5;
    int lane = threadIdx.x & 31;
    if (e >= EE) return;
    int qi = qidx[e], ni = nidx[e];
    int h = lane >> 4;
    float mx = __int_as_float(segmax[qi * 2 + h]);
    float ex = expf(scores[e * 2 + h] - mx);
    if ((lane & 15) == 0) atomicAddF32(&denom[qi * 2 + h], ex);

    const float4* vp = (const float4*)(QKV + (size_t)ni * QKVLD + 512 + lane * 8);
    float4 v0 = vp[0], v1 = vp[1];
    float* op = out_raw + (size_t)qi * CCH + lane * 8;
    atomicAddF32(op + 0, ex * v0.x);
    atomicAddF32(op + 1, ex * v0.y);
    atomicAddF32(op + 2, ex * v0.z);
    atomicAddF32(op + 3, ex * v0.w);
    atomicAddF32(op + 4, ex * v1.x);
    atomicAddF32(op + 5, ex * v1.y);
    atomicAddF32(op + 6, ex * v1.z);
    atomicAddF32(op + 7, ex * v1.w);
}

// -------------------------------------------------------- softmax denominator
__global__ void normalize_out(float* __restrict__ out_raw,
                              const float* __restrict__ denom) {
    int i = blockIdx.x * blockDim.x + threadIdx.x;
    if (i >= NPTS * CCH) return;
    int n = i >> 8;
    int h = (i >> 7) & 1;
    float d = denom[n * 2 + h];
    out_raw[i] = (d > 0.0f) ? out_raw[i] / d : 0.0f;
}

// ------------------------------------------------------- wave-per-row LayerNorm
__global__ void __launch_bounds__(256)
layernorm(const float* __restrict__ y, const float* __restrict__ gamma,
          const float* __restrict__ beta, float* __restrict__ out) {
    int r    = (blockIdx.x * blockDim.x + threadIdx.x) >> 5;
    int lane = threadIdx.x & 31;
    if (r >= NPTS) return;
    const float4* yp = (const float4*)(y + (size_t)r * CCH + lane * 8);
    float4 a = yp[0], b = yp[1];
    float s  = a.x + a.y + a.z + a.w + b.x + b.y + b.z + b.w;
    float s2 = a.x * a.x + a.y * a.y + a.z * a.z + a.w * a.w
             + b.x * b.x + b.y * b.y + b.z * b.z + b.w * b.w;
    for (int m = 1; m < 32; m <<= 1) {
        s  += __shfl_xor(s, m);
        s2 += __shfl_xor(s2, m);
    }
    float mean = s * (1.0f / CCH);
    float var  = s2 * (1.0f / CCH) - mean * mean;
    float rstd = rsqrtf(var + 1e-5f);
    int c0 = lane * 8;
    float* op = out + (size_t)r * CCH + c0;
    float vin[8] = {a.x, a.y, a.z, a.w, b.x, b.y, b.z, b.w};
#pragma unroll
    for (int j = 0; j < 8; ++j)
        op[j] = (vin[j] - mean) * rstd * gamma[c0 + j] + beta[c0 + j];
}

// ------------------------------------------------------------------- launcher
extern "C" void kernel_launch(void* const* d_in, const int* in_sizes, int n_in,
                              void* d_out, int out_size, void* d_ws, size_t ws_size,
                              hipStream_t stream) {
    const float* features = (const float*)d_in[0];
    const float* coords   = (const float*)d_in[1];
    const float* Wq       = (const float*)d_in[2];
    const float* Wk       = (const float*)d_in[3];
    const float* Wv       = (const float*)d_in[4];
    const float* Wo       = (const float*)d_in[5];
    const float* Wpos     = (const float*)d_in[6];
    const float* bpos     = (const float*)d_in[7];
    const float* gamma    = (const float*)d_in[8];
    const float* beta     = (const float*)d_in[9];
    const int*   nidx     = (const int*)d_in[10];
    const int*   qidx     = (const int*)d_in[11];
    float* out = (float*)d_out;

    float* ws      = (float*)d_ws;
    float* QKV     = ws;                                // N*768 floats
    float* out_raw = QKV + (size_t)NPTS * QKVLD;        // N*256
    float* scores  = out_raw + (size_t)NPTS * CCH;      // E*2
    float* denom   = scores + (size_t)EE * HH;          // N*2
    int*   segmax  = (int*)(denom + (size_t)NPTS * HH); // N*2
    float* y       = QKV;  // QKV dead after edge_accum; reuse for residual output

    // 1. init scratch
    init_ws<<<(NPTS * CCH + 255) / 256, 256, 0, stream>>>(out_raw, denom, segmax);

    // 2-4. Q/K/V projections (fp32 WMMA), packed into QKV rows of 768
    {
        int mtiles = NPTS / 16, ngroups = CCH / 64;          // 1250, 4
        int blocks = (mtiles * ngroups * 32 + 255) / 256;
        gemm_xt_wmma<<<blocks, 256, 0, stream>>>(features, Wq, QKV, nullptr, mtiles, ngroups, QKVLD, 0);
        gemm_xt_wmma<<<blocks, 256, 0, stream>>>(features, Wk, QKV, nullptr, mtiles, ngroups, QKVLD, 256);
        gemm_xt_wmma<<<blocks, 256, 0, stream>>>(features, Wv, QKV, nullptr, mtiles, ngroups, QKVLD, 512);
    }

    // 5-6. edge passes (one wave per edge)
    {
        int blocks = (EE * 32 + 255) / 256;
        edge_scores<<<blocks, 256, 0, stream>>>(QKV, coords, Wpos, bpos, qidx, nidx, scores, segmax);
        edge_accum<<<blocks, 256, 0, stream>>>(QKV, scores, segmax, qidx, nidx, denom, out_raw);
    }

    // 7. softmax normalization at node level
    normalize_out<<<(NPTS * CCH + 255) / 256, 256, 0, stream>>>(out_raw, denom);

    // 8. output projection + residual (fp32 WMMA)
    {
        int mtiles = NPTS / 16, ngroups = CCH / 64;
        int blocks = (mtiles * ngroups * 32 + 255) / 256;
        gemm_xt_wmma<<<blocks, 256, 0, stream>>>(out_raw, Wo, y, features, mtiles, ngroups, CCH, 0);
    }

    // 9. LayerNorm
    layernorm<<<(NPTS * 32 + 255) / 256, 256, 0, stream>>>(y, gamma, beta, out);
}